// leapLSTM_36842229465462
// MI455X (gfx1250) — compile-verified
//
#include <hip/hip_runtime.h>
#include <math.h>

// Problem constants
constexpr int BB   = 32;
constexpr int TT   = 512;
constexpr int EE   = 256;
constexpr int HH   = 512;
constexpr int RH   = 10;
constexpr int FEAT = 190;   // 3*60 + 10
constexpr int DD0  = 100;
constexpr int NROWS = BB * TT; // 16384

// LDS layout for scan kernel: 4 gates x 16 cols, padded column stride (v2f units)
constexpr int CSTR = 258;                         // 256 + 2 pad -> conflict-free b64 reads
constexpr int SCAN_LDS_BYTES = 4 * 16 * CSTR * 8; // 132096

typedef __attribute__((ext_vector_type(2))) float v2f;
typedef __attribute__((ext_vector_type(8))) float v8f;

__device__ __forceinline__ v8f wmma4(v2f a, v2f b, v8f c) {
  // fp32 WMMA: D(16x16) = A(16x4) * B(4x16) + C   (full fp32 precision)
  return __builtin_amdgcn_wmma_f32_16x16x4_f32(false, a, false, b, (short)0, c, false, false);
}

__device__ __forceinline__ float sigm(float x) { return 1.f / (1.f + __expf(-x)); }
// tanh via fast exp: 1 - 2/(e^{2x}+1); saturates correctly at +/-inf.
__device__ __forceinline__ float ftanh(float x) { return 1.f - 2.f / (__expf(2.f * x) + 1.f); }

// ---------------------------------------------------------------------------
// Pack a (Kreal x Nreal) row-major weight (leading dim ldw) into per-column
// (k,k+1) pairs: out[n*K2 + k/2] = {W[k][n], W[k+1][n]}, zero padded.
// Matches the WMMA f32 B-operand per-lane fetch (n = lane%16,
// koff = 2*(lane/16) -> one v2f load).
// ---------------------------------------------------------------------------
__global__ void pack_pairs_kernel(const float* __restrict__ W, int ldw,
                                  int Kreal, int Nreal, int K2, int Npad,
                                  v2f* __restrict__ outp) {
  int id = blockIdx.x * blockDim.x + threadIdx.x;
  int total = Npad * K2;
  if (id >= total) return;
  int n = id / K2;
  int k = (id % K2) * 2;
  v2f v;
  v.x = (n < Nreal && k     < Kreal) ? W[(size_t)k       * ldw + n] : 0.f;
  v.y = (n < Nreal && (k+1) < Kreal) ? W[(size_t)(k + 1) * ldw + n] : 0.f;
  outp[id] = v;
}

// ---------------------------------------------------------------------------
// Implicit-GEMM conv (kernel sizes 3/4/5, SAME padding).
// rows = (b,t), K = taps*256, N = 60 (padded 64). One wave per (mtile,ntile),
// 4 waves / block, blockIdx.y = conv id.
// ---------------------------------------------------------------------------
__global__ void conv_wmma_kernel(const float* __restrict__ x,
                                 const v2f* __restrict__ Wc1,
                                 const v2f* __restrict__ Wc2,
                                 const v2f* __restrict__ Wc3,
                                 const float* __restrict__ b1,
                                 const float* __restrict__ b2,
                                 const float* __restrict__ b3,
                                 float* __restrict__ fcore) {
  int wave = threadIdx.x >> 5;         // 0..3 = ntile
  int lane = threadIdx.x & 31;
  int mtile = blockIdx.x;              // 0..1023
  int conv = blockIdx.y;               // 0..2
  int taps = 3 + conv;
  int padlow = (conv == 2) ? 2 : 1;    // jax SAME: low = (k-1)//2
  const v2f* P = (conv == 0) ? Wc1 : ((conv == 1) ? Wc2 : Wc3);
  const float* bias = (conv == 0) ? b1 : ((conv == 1) ? b2 : b3);
  int K = taps * EE;
  int K2 = taps * 128;

  int row0 = mtile * 16;
  int b = row0 / TT;
  int t0 = row0 % TT;                  // 16 consecutive t, same b
  int mA = lane & 15;
  int koff = (lane >> 4) << 1;
  int n = wave * 16 + (lane & 15);

  v8f acc = {};
  for (int k = 0; k < K; k += 4) {
    int kk = k + koff;
    int tap = kk >> 8;
    int e = kk & 255;
    int ts = t0 + mA + tap - padlow;
    v2f a;
    if ((unsigned)ts < (unsigned)TT) {
      const float* px = x + ((size_t)b * TT + ts) * EE + e;
      a.x = px[0]; a.y = px[1];
    } else {
      a.x = 0.f; a.y = 0.f;
    }
    v2f bv = P[(size_t)n * K2 + (kk >> 1)];
    acc = wmma4(a, bv, acc);
  }

  if (n < 60) {
    float bval = bias[n];
    int cbase = conv * 60;
    int hi = (lane >> 4) << 3;
#pragma unroll
    for (int r = 0; r < 8; ++r) {
      int t = t0 + r + hi;             // D row M = r + 8*(lane/16)
      fcore[((size_t)b * TT + t) * FEAT + cbase + n] = acc[r] + bval;
    }
  }
}

// ---------------------------------------------------------------------------
// Reverse-LSTM x projection: zx[t][b][g] = rev_b[g] + x[b][T-1-t] . rev_Wx[:,g]
// ---------------------------------------------------------------------------
__global__ void zxrev_kernel(const float* __restrict__ x,
                             const float* __restrict__ Wx,
                             const float* __restrict__ bias,
                             float* __restrict__ zx) {
  int id = blockIdx.x * blockDim.x + threadIdx.x;
  if (id >= TT * BB * 40) return;
  int g = id % 40;
  int tb = id / 40;
  int b = tb % BB;
  int t = tb / BB;
  const float* xp = x + ((size_t)b * TT + (TT - 1 - t)) * EE;
  float acc = bias[g];
  for (int e = 0; e < EE; ++e) acc += xp[e] * Wx[e * 40 + g];
  zx[id] = acc;
}

// ---------------------------------------------------------------------------
// Reverse-LSTM recurrence (H=10): one block, 320 threads, thread = (b, j).
// Writes hidden states into fcore[...][180+j].
// ---------------------------------------------------------------------------
__global__ void revscan_kernel(const float* __restrict__ zx,
                               const float* __restrict__ Wh,
                               float* __restrict__ fcore) {
  __shared__ float hs[BB][RH];
  int tid = threadIdx.x;
  int b = tid / RH;
  int j = tid % RH;
  float wi[RH], wf[RH], wg[RH], wo[RH];
  for (int k = 0; k < RH; ++k) {
    wi[k] = Wh[k * 40 + j];
    wf[k] = Wh[k * 40 + 10 + j];
    wg[k] = Wh[k * 40 + 20 + j];
    wo[k] = Wh[k * 40 + 30 + j];
  }
  float c = 0.f;
  hs[b][j] = 0.f;
  __syncthreads();
  for (int s = 0; s < TT; ++s) {
    const float* z = zx + ((size_t)s * BB + b) * 40;
    float zi = z[j], zf = z[10 + j], zg = z[20 + j], zo = z[30 + j];
    for (int k = 0; k < RH; ++k) {
      float hk = hs[b][k];
      zi += hk * wi[k]; zf += hk * wf[k]; zg += hk * wg[k]; zo += hk * wo[k];
    }
    c = sigm(zf) * c + sigm(zi) * ftanh(zg);
    float h = sigm(zo) * ftanh(c);
    fcore[((size_t)b * TT + s) * FEAT + 180 + j] = h;
    __syncthreads();
    hs[b][j] = h;
    __syncthreads();
  }
}

// ---------------------------------------------------------------------------
// Gating hidden layer: hid = relu([ff_seq, x] @ d0_W + d0_b)
// rows = (b,t), K = 446 padded to 448, N = 100 padded to 112 (7 ntiles).
// ---------------------------------------------------------------------------
__global__ void hid_wmma_kernel(const float* __restrict__ x,
                                const float* __restrict__ fcore,
                                const v2f* __restrict__ WdP,
                                const float* __restrict__ d0b,
                                float* __restrict__ hidp) {
  int wave = threadIdx.x >> 5;        // 0..6 = ntile
  int lane = threadIdx.x & 31;
  int mtile = blockIdx.x;
  int row0 = mtile * 16;
  int b = row0 / TT;
  int t0 = row0 % TT;
  int mA = lane & 15;
  int koff = (lane >> 4) << 1;
  int n = wave * 16 + (lane & 15);
  int tA = t0 + mA;

  v8f acc = {};
  for (int k = 0; k < 448; k += 4) {
    int kk = k + koff;
    v2f a;
    if (kk < FEAT) {                       // FEAT=190 even; pairs don't straddle
      if (tA < TT - 1) {
        const float* p = fcore + ((size_t)b * TT + tA + 1) * FEAT + kk;
        a.x = p[0]; a.y = p[1];
      } else { a.x = 0.f; a.y = 0.f; }
    } else {
      int e = kk - FEAT;
      if (e < EE) {
        const float* p = x + ((size_t)b * TT + tA) * EE + e;
        a.x = p[0]; a.y = p[1];
      } else { a.x = 0.f; a.y = 0.f; }
    }
    v2f bv = WdP[(size_t)n * 224 + (kk >> 1)];
    acc = wmma4(a, bv, acc);
  }

  if (n < DD0) {
    float bval = d0b[n];
    int hi = (lane >> 4) << 3;
#pragma unroll
    for (int r = 0; r < 8; ++r) {
      int row = row0 + r + hi;
      float v = acc[r] + bval;
      hidp[(size_t)row * DD0 + n] = v > 0.f ? v : 0.f;
    }
  }
}

// ---------------------------------------------------------------------------
// pi = hid @ d1_W + d1_b ; d = softmax((pi+gn)/1e-5) -> store d[:,0]
// ---------------------------------------------------------------------------
__global__ void pid_kernel(const float* __restrict__ hidp,
                           const float* __restrict__ d1W,
                           const float* __restrict__ d1b,
                           const float* __restrict__ gn,
                           float* __restrict__ dgate) {
  int row = blockIdx.x * blockDim.x + threadIdx.x;
  if (row >= NROWS) return;
  int b = row / TT;
  int t = row % TT;
  const float* hp = hidp + (size_t)row * DD0;
  float p0 = d1b[0], p1 = d1b[1];
  for (int k = 0; k < DD0; ++k) {
    float h = hp[k];
    p0 += h * d1W[k * 2 + 0];
    p1 += h * d1W[k * 2 + 1];
  }
  const float* g = gn + ((size_t)t * BB + b) * 2;
  float a0 = p0 + g[0];
  float a1 = p1 + g[1];
  float mx = fmaxf(a0, a1);
  float e0 = __expf((a0 - mx) * 1e5f);   // /TEMP, TEMP=1e-5
  float e1 = __expf((a1 - mx) * 1e5f);
  dgate[t * BB + b] = e0 / (e0 + e1);
}

// ---------------------------------------------------------------------------
// Main LSTM scan: persistent kernel, 32 blocks x 2 waves. Block jt owns
// h-columns [jt*16, jt*16+16); wave mt owns batch rows [mt*16, mt*16+16).
// The Wh B-operand pairs for all 4 gate tiles (128 KB) are staged in LDS
// once (padded column stride -> conflict-free ds_load_b64), then reused for
// all 512 steps. Per step: z = [h_prev, x_t] @ [Wh; Wx] + b via
// v_wmma_f32_16x16x4_f32, gated state update, device-scope spin barrier.
// ---------------------------------------------------------------------------
__global__ void scan_kernel(const float* __restrict__ x,
                            const v2f* __restrict__ WhP,
                            const v2f* __restrict__ WxP,
                            const float* __restrict__ cellb,
                            const float* __restrict__ dgate,
                            float* __restrict__ hbuf,
                            float* __restrict__ out,
                            int* __restrict__ counter) {
  extern __shared__ char smem[];
  v2f* Bh = (v2f*)smem;                 // [gate][col] stride CSTR in v2f
  int tid = threadIdx.x;                // 0..63
  int lane = tid & 31;
  int mt = tid >> 5;                    // wave = batch half
  int j0 = blockIdx.x * 16;

  // Stage Wh pairs for this block's 4 gate tiles into LDS.
  for (int i = tid; i < 4 * 16 * 256; i += 64) {
    int gate = i >> 12;
    int col = (i >> 8) & 15;
    int k2 = i & 255;
    Bh[(gate * 16 + col) * CSTR + k2] =
        WhP[(size_t)(gate * HH + j0 + col) * 256 + k2];
  }
  __syncthreads();

  int mA = lane & 15;
  int koff = (lane >> 4) << 1;
  int bidx = mt * 16 + mA;              // A row (batch) for loads
  int ncol = lane & 15;
  int hi = (lane >> 4) << 3;

  int nI = j0, nF = HH + j0, nG = 2 * HH + j0, nO = 3 * HH + j0;
  float bI = cellb[nI + ncol], bF = cellb[nF + ncol];
  float bG = cellb[nG + ncol], bO = cellb[nO + ncol];
  const v2f* LI = Bh + (0 * 16 + ncol) * CSTR;
  const v2f* LF = Bh + (1 * 16 + ncol) * CSTR;
  const v2f* LG = Bh + (2 * 16 + ncol) * CSTR;
  const v2f* LO = Bh + (3 * 16 + ncol) * CSTR;
  const v2f* PXI = WxP + (size_t)(nI + ncol) * 128;
  const v2f* PXF = WxP + (size_t)(nF + ncol) * 128;
  const v2f* PXG = WxP + (size_t)(nG + ncol) * 128;
  const v2f* PXO = WxP + (size_t)(nO + ncol) * 128;

  v8f cst = {};
  v8f hst = {};

  for (int t = 0; t < TT; ++t) {
    const float* hprev = hbuf + (size_t)(t & 1) * (BB * HH);
    float* hnext = hbuf + (size_t)((t + 1) & 1) * (BB * HH);
    const float* xrow = x + ((size_t)bidx * TT + t) * EE;

    v8f aI = {}, aF = {}, aG = {}, aO = {};
    const float* hrow = hprev + (size_t)bidx * HH;
#pragma unroll 2
    for (int k = 0; k < HH; k += 4) {
      int kk = k + koff;
      v2f a; a.x = hrow[kk]; a.y = hrow[kk + 1];
      int k2 = kk >> 1;
      aI = wmma4(a, LI[k2], aI);        // B from LDS (ds_load_b64)
      aF = wmma4(a, LF[k2], aF);
      aG = wmma4(a, LG[k2], aG);
      aO = wmma4(a, LO[k2], aO);
    }
#pragma unroll 2
    for (int k = 0; k < EE; k += 4) {
      int kk = k + koff;
      v2f a; a.x = xrow[kk]; a.y = xrow[kk + 1];
      int k2 = kk >> 1;
      aI = wmma4(a, PXI[k2], aI);
      aF = wmma4(a, PXF[k2], aF);
      aG = wmma4(a, PXG[k2], aG);
      aO = wmma4(a, PXO[k2], aO);
    }

#pragma unroll
    for (int r = 0; r < 8; ++r) {
      float zi = aI[r] + bI, zf = aF[r] + bF;
      float zg = aG[r] + bG, zo = aO[r] + bO;
      float cc = sigm(zf) * cst[r] + sigm(zi) * ftanh(zg);
      float hc = sigm(zo) * ftanh(cc);
      int bglob = mt * 16 + r + hi;          // D row
      float d0 = dgate[t * BB + bglob];
      float d1 = 1.f - d0;
      float hn = d0 * hc + d1 * hst[r];
      cst[r] = d0 * cc + d1 * cst[r];
      hst[r] = hn;
      out[((size_t)bglob * TT + t) * HH + j0 + ncol] = hc;
      hnext[(size_t)bglob * HH + j0 + ncol] = hn;
    }

    // device-scope grid barrier (32 arrivals per step)
    __threadfence();
    __syncthreads();
    if (tid == 0) {
      __hip_atomic_fetch_add(counter, 1, __ATOMIC_ACQ_REL, __HIP_MEMORY_SCOPE_AGENT);
      int target = 32 * (t + 1);
      while (__hip_atomic_load(counter, __ATOMIC_ACQUIRE, __HIP_MEMORY_SCOPE_AGENT) < target) {
        __builtin_amdgcn_s_sleep(1);
      }
    }
    __syncthreads();
    __threadfence();
  }
}

// ---------------------------------------------------------------------------
extern "C" void kernel_launch(void* const* d_in, const int* in_sizes, int n_in,
                              void* d_out, int out_size, void* d_ws, size_t ws_size,
                              hipStream_t stream) {
  (void)in_sizes; (void)n_in; (void)out_size; (void)ws_size;
  const float* x      = (const float*)d_in[0];
  const float* c1w    = (const float*)d_in[1];
  const float* c1b    = (const float*)d_in[2];
  const float* c2w    = (const float*)d_in[3];
  const float* c2b    = (const float*)d_in[4];
  const float* c3w    = (const float*)d_in[5];
  const float* c3b    = (const float*)d_in[6];
  const float* revWx  = (const float*)d_in[7];
  const float* revWh  = (const float*)d_in[8];
  const float* revb   = (const float*)d_in[9];
  const float* d0W    = (const float*)d_in[10];
  const float* d0b    = (const float*)d_in[11];
  const float* d1W    = (const float*)d_in[12];
  const float* d1b    = (const float*)d_in[13];
  const float* cellWx = (const float*)d_in[14];
  const float* cellWh = (const float*)d_in[15];
  const float* cellb  = (const float*)d_in[16];
  const float* gn     = (const float*)d_in[17];
  float* out = (float*)d_out;

  char* p = (char*)d_ws;
  auto alloc = [&](size_t bytes) -> void* {
    void* r = (void*)p;
    p += (bytes + 255) & ~(size_t)255;
    return r;
  };
  float* fcore = (float*)alloc(sizeof(float) * (size_t)NROWS * FEAT);
  float* zxrev = (float*)alloc(sizeof(float) * (size_t)TT * BB * 40);
  float* hidp  = (float*)alloc(sizeof(float) * (size_t)NROWS * DD0);
  float* dgate = (float*)alloc(sizeof(float) * NROWS);
  v2f* WhP = (v2f*)alloc(sizeof(v2f) * 2048 * 256);
  v2f* WxP = (v2f*)alloc(sizeof(v2f) * 2048 * 128);
  v2f* WdP = (v2f*)alloc(sizeof(v2f) * 112 * 224);
  v2f* Wc1 = (v2f*)alloc(sizeof(v2f) * 64 * 384);
  v2f* Wc2 = (v2f*)alloc(sizeof(v2f) * 64 * 512);
  v2f* Wc3 = (v2f*)alloc(sizeof(v2f) * 64 * 640);
  float* hbuf = (float*)alloc(sizeof(float) * 2 * BB * HH);
  int* counter = (int*)alloc(256);

  hipMemsetAsync(hbuf, 0, sizeof(float) * 2 * BB * HH, stream);
  hipMemsetAsync(counter, 0, sizeof(int), stream);

  auto pk = [&](const float* W, int ldw, int Kreal, int Nreal, int K2, int Npad, v2f* o) {
    int total = Npad * K2;
    pack_pairs_kernel<<<(total + 255) / 256, 256, 0, stream>>>(W, ldw, Kreal, Nreal, K2, Npad, o);
  };
  pk(cellWh, 2048, 512, 2048, 256, 2048, WhP);
  pk(cellWx, 2048, 256, 2048, 128, 2048, WxP);
  pk(d0W, 100, 446, 100, 224, 112, WdP);
  pk(c1w, 60, 3 * 256, 60, 384, 64, Wc1);
  pk(c2w, 60, 4 * 256, 60, 512, 64, Wc2);
  pk(c3w, 60, 5 * 256, 60, 640, 64, Wc3);

  zxrev_kernel<<<(TT * BB * 40 + 255) / 256, 256, 0, stream>>>(x, revWx, revb, zxrev);
  conv_wmma_kernel<<<dim3(NROWS / 16, 3), 128, 0, stream>>>(x, Wc1, Wc2, Wc3, c1b, c2b, c3b, fcore);
  revscan_kernel<<<1, BB * RH, 0, stream>>>(zxrev, revWh, fcore);
  hid_wmma_kernel<<<NROWS / 16, 224, 0, stream>>>(x, fcore, WdP, d0b, hidp);
  pid_kernel<<<NROWS / 256, 256, 0, stream>>>(hidp, d1W, d1b, gn, dgate);
  scan_kernel<<<32, 64, SCAN_LDS_BYTES, stream>>>(x, WhP, WxP, cellb, dgate, hbuf, out, counter);
}